// EncoderGNN_65335042506960
// MI455X (gfx1250) — compile-verified
//
#include <hip/hip_runtime.h>

#define DCH 128
#define LN_EPS 1e-5f

typedef float v2f __attribute__((ext_vector_type(2)));
typedef float v8f __attribute__((ext_vector_type(8)));

// ---------------------------------------------------------------- degree pass
__global__ void zero_cnt_kernel(unsigned* __restrict__ cnt, int n) {
  int i = blockIdx.x * blockDim.x + threadIdx.x;
  if (i < n) cnt[i] = 0u;
}

__global__ void count_edges_kernel(const int* __restrict__ dst,
                                   unsigned* __restrict__ cnt, int e) {
  int i = blockIdx.x * blockDim.x + threadIdx.x;
  int stride = gridDim.x * blockDim.x;
  for (; i < e; i += stride) atomicAdd(&cnt[dst[i]], 1u);
}

__global__ void dinv_kernel(const unsigned* __restrict__ cnt,
                            float* __restrict__ dinv, int n) {
  int i = blockIdx.x * blockDim.x + threadIdx.x;
  if (i < n) dinv[i] = rsqrtf(1.0f + (float)cnt[i]);
}

// coef[e] = dinv[src]*dinv[dst], computed once, reused by all 3 layers
__global__ void coef_kernel(const int* __restrict__ src,
                            const int* __restrict__ dst,
                            const float* __restrict__ dinv,
                            float* __restrict__ coef, int e) {
  int i = blockIdx.x * blockDim.x + threadIdx.x;
  int stride = gridDim.x * blockDim.x;
  for (; i < e; i += stride) coef[i] = dinv[src[i]] * dinv[dst[i]];
}

// ------------------------------------------------- fp32 WMMA GEMM: out = X @ W
// One 256-thread block = 8 waves. Block handles 16 rows (one M tile, staged in
// LDS), each wave owns one 16-wide N tile. K loop: 32 x v_wmma_f32_16x16x4_f32.
__global__ __launch_bounds__(256)
void gemm_wmma_kernel(const float* __restrict__ X, const float* __restrict__ W,
                      float* __restrict__ out, int nrows) {
  __shared__ float xs[16 * DCH];          // 8 KB A-tile, shared by 8 waves
  const int m0 = blockIdx.x * 16;
  const int t  = threadIdx.x;             // 0..255

  // Cooperative LDS fill: the 16x128 row-major tile is 2048 contiguous floats.
  const int tile_elems = (nrows - m0 >= 16 ? 16 : (nrows - m0)) * DCH;
  const float4* src4 = (const float4*)(X + (size_t)m0 * DCH);
  float4* lds4 = (float4*)xs;
  {
    float4 z; z.x = z.y = z.z = z.w = 0.0f;
    int i0 = t, i1 = t + 256;             // 512 float4 = 2048 floats
    lds4[i0] = (i0 * 4 < tile_elems) ? src4[i0] : z;
    lds4[i1] = (i1 * 4 < tile_elems) ? src4[i1] : z;
  }
  __syncthreads();

  const int wave = t >> 5;                // 0..7  -> N tile
  const int lane = t & 31;
  const int n0   = wave * 16;
  const int row  = lane & 15;             // A: M index
  const int col  = lane & 15;             // B/C: N index
  const int koff = (lane >> 4) * 2;       // half-wave K pair select
  const int half = lane >> 4;

  v8f c = {};                             // 16x16 f32 accumulator (8 VGPRs)
#pragma unroll 8
  for (int k0 = 0; k0 < DCH; k0 += 4) {
    v2f a, b;
    a.x = xs[row * DCH + k0 + koff];
    a.y = xs[row * DCH + k0 + koff + 1];
    b.x = W[(k0 + koff)     * DCH + n0 + col];
    b.y = W[(k0 + koff + 1) * DCH + n0 + col];
    c = __builtin_amdgcn_wmma_f32_16x16x4_f32(
        /*neg_a=*/false, a, /*neg_b=*/false, b,
        /*c_mod=*/(short)0, c, /*reuse_a=*/false, /*reuse_b=*/false);
  }

  // C/D layout: VGPR v -> rows v (lanes 0-15) and v+8 (lanes 16-31).
  float* obase = out + (size_t)m0 * DCH + n0;
  if (m0 + 16 <= nrows) {                 // block-uniform fast path: no exec churn
#pragma unroll
    for (int v = 0; v < 8; ++v) {
      int m = v + half * 8;
      obase[(size_t)m * DCH + col] = c[v];
    }
  } else {                                // tail tile (not taken for N%16==0)
#pragma unroll
    for (int v = 0; v < 8; ++v) {
      int m = v + half * 8;
      if (m0 + m < nrows) obase[(size_t)m * DCH + col] = c[v];
    }
  }
}

// ------------------------------- agg = hlin * dinv^2 + bias (self-loop fused)
__global__ void init_agg_kernel(const float* __restrict__ hlin,
                                const float* __restrict__ dinv,
                                const float* __restrict__ bias,
                                float* __restrict__ agg, int n) {
  int i4 = blockIdx.x * blockDim.x + threadIdx.x;   // over n*32 float4s
  if (i4 >= n * 32) return;
  int node = i4 >> 5, c4 = i4 & 31;
  float dv = dinv[node];
  float sc = dv * dv;
  float4 h = ((const float4*)hlin)[i4];
  float4 b = ((const float4*)bias)[c4];
  float4 o;
  o.x = h.x * sc + b.x;
  o.y = h.y * sc + b.y;
  o.z = h.z * sc + b.z;
  o.w = h.w * sc + b.w;
  ((float4*)agg)[i4] = o;
}

// ----------------------- wave-per-edge gather (L2-resident) + f32 scatter-add
__global__ __launch_bounds__(256)
void edge_scatter_kernel(const float* __restrict__ hlin,
                         const float* __restrict__ coef,
                         const int* __restrict__ src,
                         const int* __restrict__ dst,
                         float* __restrict__ agg, int e) {
  int gtid = blockIdx.x * blockDim.x + threadIdx.x;
  int wid  = gtid >> 5;
  int lane = threadIdx.x & 31;
  int nw   = (gridDim.x * blockDim.x) >> 5;
  for (int ed = wid; ed < e; ed += nw) {
    int s = src[ed], d = dst[ed];
    float cf = coef[ed];
    int nxt = ed + nw;
    if (nxt < e) {                        // global_prefetch_b8 of next src row
      int ns = src[nxt];
      __builtin_prefetch(hlin + (size_t)ns * DCH + lane * 4, 0, 1);
    }
    float4 v = ((const float4*)(hlin + (size_t)s * DCH))[lane];
    float* ap = agg + (size_t)d * DCH + lane * 4;
    unsafeAtomicAdd(ap + 0, v.x * cf);    // global_atomic_add_f32 (L2)
    unsafeAtomicAdd(ap + 1, v.y * cf);
    unsafeAtomicAdd(ap + 2, v.z * cf);
    unsafeAtomicAdd(ap + 3, v.w * cf);
  }
}

// -------------- wave-per-node LayerNorm + ReLU + JK running max (all fused)
__global__ __launch_bounds__(256)
void post_kernel(const float* __restrict__ agg, const float* __restrict__ gamma,
                 const float* __restrict__ beta, float* __restrict__ hout,
                 float* __restrict__ jk, int do_relu, int first, int n) {
  int wid  = (blockIdx.x * blockDim.x + threadIdx.x) >> 5;
  int lane = threadIdx.x & 31;
  if (wid >= n) return;

  float4 v = ((const float4*)(agg + (size_t)wid * DCH))[lane];

  float s = v.x + v.y + v.z + v.w;
#pragma unroll
  for (int off = 16; off > 0; off >>= 1) s += __shfl_xor(s, off, 32);
  float mu = s * (1.0f / DCH);

  float dx = v.x - mu, dy = v.y - mu, dz = v.z - mu, dw = v.w - mu;
  float q = dx * dx + dy * dy + dz * dz + dw * dw;
#pragma unroll
  for (int off = 16; off > 0; off >>= 1) q += __shfl_xor(q, off, 32);
  float inv = rsqrtf(q * (1.0f / DCH) + LN_EPS);

  float4 g = ((const float4*)gamma)[lane];
  float4 b = ((const float4*)beta)[lane];
  float4 y;
  y.x = dx * inv * g.x + b.x;
  y.y = dy * inv * g.y + b.y;
  y.z = dz * inv * g.z + b.z;
  y.w = dw * inv * g.w + b.w;
  if (do_relu) {
    y.x = fmaxf(y.x, 0.0f); y.y = fmaxf(y.y, 0.0f);
    y.z = fmaxf(y.z, 0.0f); y.w = fmaxf(y.w, 0.0f);
  }

  ((float4*)(hout + (size_t)wid * DCH))[lane] = y;

  float4* op = (float4*)(jk + (size_t)wid * DCH) + lane;
  if (first) {
    *op = y;
  } else {
    float4 o = *op;
    o.x = fmaxf(o.x, y.x); o.y = fmaxf(o.y, y.y);
    o.z = fmaxf(o.z, y.z); o.w = fmaxf(o.w, y.w);
    *op = o;
  }
}

// ---------------------------------------------------------------------- driver
extern "C" void kernel_launch(void* const* d_in, const int* in_sizes, int n_in,
                              void* d_out, int out_size, void* d_ws, size_t ws_size,
                              hipStream_t stream) {
  const int N = in_sizes[0] / DCH;      // 100000
  const int E = in_sizes[1] / 2;        // 1600000

  const float* feat = (const float*)d_in[0];
  const int*   ei   = (const int*)d_in[1];
  const int*   src  = ei;
  const int*   dst  = ei + E;
  const float* Wm[3] = {(const float*)d_in[2], (const float*)d_in[6],  (const float*)d_in[10]};
  const float* bi[3] = {(const float*)d_in[3], (const float*)d_in[7],  (const float*)d_in[11]};
  const float* gm[3] = {(const float*)d_in[4], (const float*)d_in[8],  (const float*)d_in[12]};
  const float* be[3] = {(const float*)d_in[5], (const float*)d_in[9],  (const float*)d_in[13]};
  float* out = (float*)d_out;

  // workspace carve-up (all 16B-aligned): cnt | dinv | hlin | agg | hbuf | coef
  float*    wsf  = (float*)d_ws;
  unsigned* cnt  = (unsigned*)wsf;
  float*    dinv = wsf + N;
  float*    hlin = wsf + 2 * (size_t)N;
  float*    agg  = hlin + (size_t)N * DCH;
  float*    hbuf = agg  + (size_t)N * DCH;
  float*    coef = hbuf + (size_t)N * DCH;

  const int nodeBlocks = (N + 255) / 256;
  zero_cnt_kernel<<<nodeBlocks, 256, 0, stream>>>(cnt, N);
  count_edges_kernel<<<2048, 256, 0, stream>>>(dst, cnt, E);
  dinv_kernel<<<nodeBlocks, 256, 0, stream>>>(cnt, dinv, N);
  coef_kernel<<<2048, 256, 0, stream>>>(src, dst, dinv, coef, E);

  const int gemmBlocks = (N + 15) / 16;          // one 16-row M tile per block
  const int elemBlocks = (N * 32 + 255) / 256;   // float4 granularity
  const int waveBlocks = (N + 7) / 8;            // 8 waves/block, wave-per-node

  const float* X = feat;
  for (int l = 0; l < 3; ++l) {
    gemm_wmma_kernel<<<gemmBlocks, 256, 0, stream>>>(X, Wm[l], hlin, N);
    init_agg_kernel<<<elemBlocks, 256, 0, stream>>>(hlin, dinv, bi[l], agg, N);
    edge_scatter_kernel<<<8192, 256, 0, stream>>>(hlin, coef, src, dst, agg, E);
    post_kernel<<<waveBlocks, 256, 0, stream>>>(agg, gm[l], be[l], hbuf, out,
                                                /*do_relu=*/(l < 2),
                                                /*first=*/(l == 0), N);
    X = hbuf;
  }
}